// DeformConv_62680752718274
// MI455X (gfx1250) — compile-verified
//
#include <hip/hip_runtime.h>

// ---------------------------------------------------------------------------
// Deformable conv (B=4, C=64, H=W=96, G=4, K=3) for gfx1250 (CDNA5, wave32).
// Both GEMM stages run on V_WMMA_F32_16X16X4_F32 (fp32 in/out, matches ref).
// ---------------------------------------------------------------------------

typedef __attribute__((ext_vector_type(2))) float v2f;
typedef __attribute__((ext_vector_type(8))) float v8f;

#define DC_B   4
#define DC_C   64
#define DC_H   96
#define DC_W   96
#define DC_HW  (DC_H * DC_W)          // 9216 pixels per image plane
#define DC_G   4
#define DC_KK  9
#define DC_NPIX DC_HW
#define DC_TILES (DC_NPIX / 16)       // 576 16-pixel tiles per image

// ===========================================================================
// Stage 1: offset-predicting conv, 64 -> 18 channels, 3x3, pad 1.
// GEMM: M = 18 (two 16-row tiles), K = 576 (kk-outer, c-inner), N = pixels.
// One wave = one 16x16 output tile. Block = 4 waves sharing (mtile, b).
// ===========================================================================
__global__ __launch_bounds__(128) void offset_conv_wmma(
    const float* __restrict__ x, const float* __restrict__ w_off,
    const float* __restrict__ b_off, float* __restrict__ offs)
{
    __shared__ float ldsA[144 * 64];          // A fragments: [step][lane][2]

    const int tid   = threadIdx.x;
    const int rest  = blockIdx.x / 144;       // = mtile*4 + b
    const int b     = rest & 3;
    const int mtile = rest >> 2;
    const int pt    = (blockIdx.x % 144) * 4 + (tid >> 5);
    const int lane  = tid & 31;
    const int half  = (lane >> 4) & 1;        // lanes 16..31 -> K offset +2, M +8
    const int ofs2  = half * 2;
    const int n     = lane & 15;              // output pixel column within tile

    // Stage the A (weight) fragments into LDS in WMMA lane layout.
    // K = kk*64 + c  ->  step s covers K = 4s..4s+3 with kk = s>>4 constant.
    for (int i = 0; i < 72; ++i) {
        int e  = tid + i * 128;               // 128*72 == 144*64 exactly
        int s  = e >> 6;
        int j  = e & 63;
        int L  = j >> 1;
        int hb = j & 1;
        int m  = L & 15;
        int c  = ((s & 15) << 2) + ((L >> 4) << 1) + hb;
        int kk = s >> 4;
        int oc = mtile * 16 + m;
        if (oc > 17) oc = 17;                 // clamp: rows >=18 never stored
        ldsA[e] = w_off[(oc * DC_C + c) * DC_KK + kk];
    }
    __syncthreads();

    // Per-lane 3x3 patch taps for this pixel (zero pad folded into mask).
    const int p = pt * 16 + n;
    const int h = p / DC_W, w = p % DC_W;
    const float* xb = x + (size_t)b * DC_C * DC_HW;
    int   idx[DC_KK];
    float msk[DC_KK];
#pragma unroll
    for (int kk = 0; kk < DC_KK; ++kk) {
        int iy = h + kk / 3 - 1;
        int ix = w + kk % 3 - 1;
        bool v = (iy >= 0) && (iy < DC_H) && (ix >= 0) && (ix < DC_W);
        int cy = iy < 0 ? 0 : (iy > DC_H - 1 ? DC_H - 1 : iy);
        int cx = ix < 0 ? 0 : (ix > DC_W - 1 ? DC_W - 1 : ix);
        idx[kk] = cy * DC_W + cx;
        msk[kk] = v ? 1.0f : 0.0f;
    }

    v8f acc = {};
#pragma unroll
    for (int s = 0; s < 144; ++s) {
        const int kk = s >> 4;                       // compile-time after unroll
        const int c0 = ((s & 15) << 2) + ofs2;       // runtime channel (addr only)
        v2f a = *(const v2f*)&ldsA[(s << 6) + (lane << 1)];
        const float* xp0 = xb + (size_t)c0 * DC_HW;
        const float* xp1 = xp0 + DC_HW;
        v2f bf;
        bf.x = msk[kk] * xp0[idx[kk]];
        bf.y = msk[kk] * xp1[idx[kk]];
        acc = __builtin_amdgcn_wmma_f32_16x16x4_f32(
            false, a, false, bf, (short)0, acc, false, false);
    }

    // Store valid rows (+bias). D row = r + 8*half, col = lane&15.
    const int base = (b * 18) * DC_NPIX + p;
#pragma unroll
    for (int r = 0; r < 8; ++r) {
        int M  = r + half * 8;
        int oc = mtile * 16 + M;
        if (oc < 18) offs[base + oc * DC_NPIX] = acc[r] + b_off[oc];
    }
}

// ===========================================================================
// Stage 2: bilinear sampling + grouped 9-tap contraction.
// Per group GEMM: M = 16 (exact tile), K = 144 (kk-outer, c-inner), N = pixels.
// One wave = one 16x16 output tile. Block = 4 waves sharing (b, g).
// ===========================================================================
__global__ __launch_bounds__(128) void deform_conv_wmma(
    const float* __restrict__ x, const float* __restrict__ w_def,
    const float* __restrict__ offs, float* __restrict__ out)
{
    __shared__ float ldsA[36 * 64];

    const int tid  = threadIdx.x;
    const int rest = blockIdx.x / 144;        // = b*4 + g
    const int b    = rest >> 2;
    const int g    = rest & 3;
    const int pt   = (blockIdx.x % 144) * 4 + (tid >> 5);
    const int lane = tid & 31;
    const int half = (lane >> 4) & 1;
    const int ofs2 = half * 2;
    const int n    = lane & 15;

    // Stage group-g weight fragments: A[m][kk*16+c] = w_def[g*16+m, c, kk].
    for (int i = 0; i < 18; ++i) {
        int e  = tid + i * 128;               // 128*18 == 36*64 exactly
        int s  = e >> 6;
        int j  = e & 63;
        int L  = j >> 1;
        int hb = j & 1;
        int m  = L & 15;
        int c  = ((s & 3) << 2) + ((L >> 4) << 1) + hb;
        int kk = s >> 2;
        ldsA[e] = w_def[((g * 16 + m) * 16 + c) * DC_KK + kk];
    }
    __syncthreads();

    // Per-lane bilinear taps for this pixel: 9 taps x 4 corners.
    // Zero-pad validity folded into the corner weights; indices clamped.
    const int p = pt * 16 + n;
    const int h = p / DC_W, w = p % DC_W;
    const float* xb = x + ((size_t)b * DC_C + g * 16) * DC_HW;
    const float* ob = offs + (size_t)b * 18 * DC_NPIX + p;

    int   i00[DC_KK], i01[DC_KK], i10[DC_KK], i11[DC_KK];
    float f00[DC_KK], f01[DC_KK], f10[DC_KK], f11[DC_KK];
#pragma unroll
    for (int kk = 0; kk < DC_KK; ++kk) {
        float dy = ob[(2 * kk) * DC_NPIX];
        float dx = ob[(2 * kk + 1) * DC_NPIX];
        float ys = (float)(h + kk / 3 - 1) + dy;
        float xs = (float)(w + kk % 3 - 1) + dx;
        float y0f = floorf(ys), x0f = floorf(xs);
        float wy1 = ys - y0f, wx1 = xs - x0f;
        float wy0 = 1.0f - wy1, wx0 = 1.0f - wx1;
        int y0 = (int)y0f, x0 = (int)x0f;
        int y1 = y0 + 1,  x1 = x0 + 1;
        float vy0 = (y0 >= 0 && y0 < DC_H) ? 1.0f : 0.0f;
        float vy1 = (y1 >= 0 && y1 < DC_H) ? 1.0f : 0.0f;
        float vx0 = (x0 >= 0 && x0 < DC_W) ? 1.0f : 0.0f;
        float vx1 = (x1 >= 0 && x1 < DC_W) ? 1.0f : 0.0f;
        int cy0 = y0 < 0 ? 0 : (y0 > DC_H - 1 ? DC_H - 1 : y0);
        int cy1 = y1 < 0 ? 0 : (y1 > DC_H - 1 ? DC_H - 1 : y1);
        int cx0 = x0 < 0 ? 0 : (x0 > DC_W - 1 ? DC_W - 1 : x0);
        int cx1 = x1 < 0 ? 0 : (x1 > DC_W - 1 ? DC_W - 1 : x1);
        i00[kk] = cy0 * DC_W + cx0;  i01[kk] = cy0 * DC_W + cx1;
        i10[kk] = cy1 * DC_W + cx0;  i11[kk] = cy1 * DC_W + cx1;
        f00[kk] = wy0 * wx0 * vy0 * vx0;  f01[kk] = wy0 * wx1 * vy0 * vx1;
        f10[kk] = wy1 * wx0 * vy1 * vx0;  f11[kk] = wy1 * wx1 * vy1 * vx1;
    }

    v8f acc = {};
#pragma unroll
    for (int s = 0; s < 36; ++s) {
        const int kk = s >> 2;                      // compile-time after unroll
        const int c0 = ((s & 3) << 2) + ofs2;       // runtime channel (addr only)
        v2f a = *(const v2f*)&ldsA[(s << 6) + (lane << 1)];
        const float* xp0 = xb + (size_t)c0 * DC_HW;
        const float* xp1 = xp0 + DC_HW;
        v2f bf;
        bf.x = f00[kk] * xp0[i00[kk]] + f01[kk] * xp0[i01[kk]]
             + f10[kk] * xp0[i10[kk]] + f11[kk] * xp0[i11[kk]];
        bf.y = f00[kk] * xp1[i00[kk]] + f01[kk] * xp1[i01[kk]]
             + f10[kk] * xp1[i10[kk]] + f11[kk] * xp1[i11[kk]];
        acc = __builtin_amdgcn_wmma_f32_16x16x4_f32(
            false, a, false, bf, (short)0, acc, false, false);
    }

#pragma unroll
    for (int r = 0; r < 8; ++r) {
        int M = r + half * 8;
        out[((size_t)(b * DC_C + g * 16 + M)) * DC_NPIX + p] = acc[r];
    }
}

// ===========================================================================
extern "C" void kernel_launch(void* const* d_in, const int* in_sizes, int n_in,
                              void* d_out, int out_size, void* d_ws, size_t ws_size,
                              hipStream_t stream) {
    const float* x     = (const float*)d_in[0];  // [4,64,96,96]
    const float* w_off = (const float*)d_in[1];  // [18,64,3,3]
    const float* b_off = (const float*)d_in[2];  // [18]
    const float* w_def = (const float*)d_in[3];  // [64,16,3,3]
    float* out  = (float*)d_out;                 // [4,64,96,96]
    float* offs = (float*)d_ws;                  // scratch: [4,18,9216] fp32 = 2.65 MB

    // Stage 1: 2 M-tiles x 4 batches x 144 blocks (4 waves/block, 576 tiles/img)
    offset_conv_wmma<<<dim3(1152), dim3(128), 0, stream>>>(x, w_off, b_off, offs);
    // Stage 2: 4 batches x 4 groups x 144 blocks
    deform_conv_wmma<<<dim3(2304), dim3(128), 0, stream>>>(x, w_def, offs, out);
}